// ImprovedGraphAutoEncoder_80023830659732
// MI455X (gfx1250) — compile-verified
//
#include <hip/hip_runtime.h>
#include <math.h>

// Problem constants (match reference)
#define BB 4
#define NN 20000          // multiple of 16 -> no row guards needed in GEMM
#define EE 320000
#define INDIM 12
#define OUTDIM 3
#define HIDD 64
#define NEG_SLOPE 0.2f
#define ALPHA_C 0.1f

#define CDIV(a,b) (((a)+(b)-1)/(b))

typedef __attribute__((ext_vector_type(2))) float v2f;
typedef __attribute__((ext_vector_type(8))) float v8f;

// ---------------------------------------------------------------------------
// WMMA GEMM, compile-time shapes: C[n x DOUT] = act(A[n x DIN] @ W + bias)
// One wave32 computes a 16-row x DOUT strip via V_WMMA_F32_16X16X4_F32.
// W (zero-padded) + bias are staged in LDS once per block (4 waves share).
// Requires n % 16 == 0 (true: n == NN == 20000).
// ---------------------------------------------------------------------------
template<int DIN, int DOUT, int ACT>
__global__ void __launch_bounds__(128)
gemm_kernel(const float* __restrict__ A, const float* __restrict__ W,
            const float* __restrict__ bias, float* __restrict__ C, int nrows)
{
    constexpr int DINP  = (DIN + 3) & ~3;     // K padded to multiple of 4
    constexpr int NT    = (DOUT + 15) / 16;   // 16-col tiles
    constexpr int DOUTP = NT * 16;            // cols padded to multiple of 16

    __shared__ float lw[DOUTP * DINP];        // W^T, K-contiguous, zero padded
    __shared__ float lb[DOUTP];

    const int tid = threadIdx.y * 32 + threadIdx.x;
    for (int idx = tid; idx < DOUTP * DINP; idx += 128) {
        const int k  = idx % DINP;
        const int cc = idx / DINP;
        lw[idx] = (k < DIN && cc < DOUT) ? W[k * DOUT + cc] : 0.f;
    }
    for (int idx = tid; idx < DOUTP; idx += 128)
        lb[idx] = (idx < DOUT) ? bias[idx] : 0.f;
    __syncthreads();

    const int rowTile = blockIdx.x * blockDim.y + threadIdx.y; // wave-uniform
    const int row0 = rowTile * 16;
    if (row0 >= nrows) return;                 // wave-uniform exit

    const int lane = threadIdx.x;              // 0..31
    const int half = lane >> 4;                // 0: lanes 0-15, 1: 16-31
    const int l16  = lane & 15;
    const int r    = row0 + l16;               // A-fragment row (always valid)

    v8f acc[NT];
    const v8f vzero = {0.f,0.f,0.f,0.f,0.f,0.f,0.f,0.f};
#pragma unroll
    for (int t = 0; t < NT; ++t) acc[t] = vzero;

#pragma unroll
    for (int k0 = 0; k0 < DINP; k0 += 4) {
        const int ka = k0 + 2 * half;          // this lane's K pair
        v2f a;
        if constexpr (DIN % 4 == 0) {
            const float2 av = *reinterpret_cast<const float2*>(
                A + (size_t)r * DIN + ka);     // 8B-aligned, unconditional
            a.x = av.x; a.y = av.y;
        } else {
            // clamped unconditional loads + select (no exec branches)
            const int base = r * DIN;
            const float ax = A[base + (ka     < DIN ? ka     : DIN - 1)];
            const float ay = A[base + (ka + 1 < DIN ? ka + 1 : DIN - 1)];
            a.x = (ka     < DIN) ? ax : 0.f;
            a.y = (ka + 1 < DIN) ? ay : 0.f;
        }
#pragma unroll
        for (int t = 0; t < NT; ++t) {
            const int c = t * 16 + l16;
            const float2 bv = *reinterpret_cast<const float2*>(
                &lw[c * DINP + ka]);           // ds_load_b64
            v2f b; b.x = bv.x; b.y = bv.y;
            acc[t] = __builtin_amdgcn_wmma_f32_16x16x4_f32(
                         false, a, false, b, (short)0, acc[t], false, false);
        }
    }

    // D layout: VGPR v holds row M = v + 8*half, col N = l16 (per tile)
#pragma unroll
    for (int t = 0; t < NT; ++t) {
        const int c = t * 16 + l16;
        if (DOUT % 16 == 0 || c < DOUT) {
            const float bv = lb[c];
#pragma unroll
            for (int v = 0; v < 8; ++v) {
                const int rr = row0 + v + 8 * half;
                float val = acc[t][v] + bv;
                if (ACT) val = fmaxf(val, 0.f);
                C[(size_t)rr * DOUT + c] = val;
            }
        }
    }
}

template<int DIN, int DOUT, int ACT>
static inline void launch_gemm(const float* A, const float* W, const float* b,
                               float* C, int n, hipStream_t stream)
{
    dim3 blk(32, 4);
    dim3 grd(CDIV(n, 16 * 4));
    gemm_kernel<DIN, DOUT, ACT><<<grd, blk, 0, stream>>>(A, W, b, C, n);
}

// ---------------------------------------------------------------------------
// Utility kernels
// ---------------------------------------------------------------------------
__global__ void fill_kernel(float* __restrict__ p, float v, int n)
{
    int i = blockIdx.x * blockDim.x + threadIdx.x;
    if (i < n) p[i] = v;
}

__global__ void extract_h0(const float* __restrict__ z, float* __restrict__ h0)
{
    int n = blockIdx.x * blockDim.x + threadIdx.x;
    if (n < NN) h0[n] = z[n * 3 + 2];
}

__global__ void comb_kernel(const float* __restrict__ g,
                            const float* __restrict__ sk,
                            float* __restrict__ out, int n)
{
    int i = blockIdx.x * blockDim.x + threadIdx.x;
    if (i < n) out[i] = g[i] + ALPHA_C * sk[i];
}

// eattr per edge + sum(eattr) for the self-loop mean
__global__ void eattr_kernel(const float* __restrict__ z,
                             const int* __restrict__ src,
                             const int* __restrict__ dst,
                             float* __restrict__ eout,
                             float* __restrict__ meansum)
{
    int e = blockIdx.x * blockDim.x + threadIdx.x;
    float val = 0.f;
    if (e < EE) {
        int s = src[e], d = dst[e];
        float d0 = z[s * 3 + 0] - z[d * 3 + 0];
        float d1 = z[s * 3 + 1] - z[d * 3 + 1];
        float d2 = z[s * 3 + 2] - z[d * 3 + 2];
        float ea = sqrtf(d0 * d0 + d1 * d1 + d2 * d2 + 1e-12f);
        eout[e] = ea;
        val = ea;
    }
    for (int o = 16; o > 0; o >>= 1) val += __shfl_down(val, o, 32);
    if ((threadIdx.x & 31) == 0) atomicAdd(meansum, val);
}

__device__ inline void atomicMaxF(float* addr, float v)
{
    if (v >= 0.f) atomicMax((int*)addr, __float_as_int(v));
    else          atomicMin((unsigned int*)addr, __float_as_uint(v));
}

__device__ inline float lrelu(float m) { return (m >= 0.f) ? m : NEG_SLOPE * m; }

// GATv2 attention score per edge slot (E real edges + N self loops)
template<int DOUT>
__global__ void gat_score(const float* __restrict__ xl,
                          const float* __restrict__ xr,
                          const int* __restrict__ src,
                          const int* __restrict__ dst,
                          const float* __restrict__ eattr,
                          const float* __restrict__ meansum,
                          const float* __restrict__ We,
                          const float* __restrict__ att,
                          float* __restrict__ ebuf,
                          float* __restrict__ emax)
{
    int f = blockIdx.x * blockDim.x + threadIdx.x;
    if (f >= EE + NN) return;
    int s, d; float ea;
    if (f < EE) { s = src[f]; d = dst[f]; ea = eattr[f]; }
    else        { s = d = f - EE; ea = meansum[0] * (1.f / (float)EE); }
    float e = 0.f;
    if constexpr (DOUT % 4 == 0) {
        const float4* pl = (const float4*)(xl + (size_t)s * DOUT);
        const float4* pr = (const float4*)(xr + (size_t)d * DOUT);
        const float4* pw = (const float4*)We;
        const float4* pt = (const float4*)att;
#pragma unroll
        for (int j = 0; j < DOUT / 4; ++j) {
            const float4 a = pl[j], b = pr[j], w = pw[j], t = pt[j];
            e += lrelu(a.x + b.x + ea * w.x) * t.x;
            e += lrelu(a.y + b.y + ea * w.y) * t.y;
            e += lrelu(a.z + b.z + ea * w.z) * t.z;
            e += lrelu(a.w + b.w + ea * w.w) * t.w;
        }
    } else {
#pragma unroll
        for (int j = 0; j < DOUT; ++j)
            e += lrelu(xl[(size_t)s * DOUT + j] + xr[(size_t)d * DOUT + j]
                       + ea * We[j]) * att[j];
    }
    ebuf[f] = e;
    atomicMaxF(&emax[d], e);
}

// w = exp(e - emax[dst]); denom[dst] += w
__global__ void gat_weight(const int* __restrict__ dst,
                           float* __restrict__ ebuf,
                           const float* __restrict__ emax,
                           float* __restrict__ den)
{
    int f = blockIdx.x * blockDim.x + threadIdx.x;
    if (f >= EE + NN) return;
    int d = (f < EE) ? dst[f] : f - EE;
    float w = expf(ebuf[f] - emax[d]);
    ebuf[f] = w;
    atomicAdd(&den[d], w);
}

// out[dst] += (w/denom[dst]) * xl[src]
template<int DOUT>
__global__ void gat_accum(const float* __restrict__ xl,
                          const int* __restrict__ src,
                          const int* __restrict__ dst,
                          const float* __restrict__ ebuf,
                          const float* __restrict__ den,
                          float* __restrict__ out)
{
    int f = blockIdx.x * blockDim.x + threadIdx.x;
    if (f >= EE + NN) return;
    int s, d;
    if (f < EE) { s = src[f]; d = dst[f]; }
    else        { s = d = f - EE; }
    const float a = ebuf[f] / den[d];
    if constexpr (DOUT % 4 == 0) {
        const float4* pl = (const float4*)(xl + (size_t)s * DOUT);
        float* po = out + (size_t)d * DOUT;
#pragma unroll
        for (int j = 0; j < DOUT / 4; ++j) {
            const float4 v = pl[j];
            atomicAdd(po + 4 * j + 0, a * v.x);
            atomicAdd(po + 4 * j + 1, a * v.y);
            atomicAdd(po + 4 * j + 2, a * v.z);
            atomicAdd(po + 4 * j + 3, a * v.w);
        }
    } else {
#pragma unroll
        for (int j = 0; j < DOUT; ++j)
            atomicAdd(&out[(size_t)d * DOUT + j], a * xl[(size_t)s * DOUT + j]);
    }
}

__global__ void bias_act_kernel(float* __restrict__ h,
                                const float* __restrict__ bias,
                                int n, int dout, int relu)
{
    int i = blockIdx.x * blockDim.x + threadIdx.x;
    if (i >= n * dout) return;
    float v = h[i] + bias[i % dout];
    if (relu) v = fmaxf(v, 0.f);
    h[i] = v;
}

// ---------------------------------------------------------------------------
// Host launcher
// ---------------------------------------------------------------------------
extern "C" void kernel_launch(void* const* d_in, const int* in_sizes, int n_in,
                              void* d_out, int out_size, void* d_ws, size_t ws_size,
                              hipStream_t stream)
{
    (void)n_in; (void)out_size; (void)ws_size;
    const float* x  = (const float*)d_in[0];
    const int*   ei = (const int*)d_in[1];
    auto P = [&](int i) { return (const float*)d_in[i]; };

    // Param layout: enc leaves always at [2..7]. GAT dicts: detect insertion
    // order (Wl,bl,Wr,br,We,att,bias) vs jax-alphabetical (We,Wl,Wr,att,bias,bl,br)
    // via g2.Wl (64*64 elements) at index 15.
    const bool insertion = (in_sizes[15] == HIDD * HIDD);
    int oWl, obl, oWr, obr, oWe, oatt, obias, skips, heads;
    if (insertion) { oWl=0; obl=1; oWr=2; obr=3; oWe=4; oatt=5; obias=6; skips=29; heads=35; }
    else           { oWe=0; oWl=1; oWr=2; oatt=3; obias=4; obl=5; obr=6; heads=29; skips=35; }
    const int g1s = 8, g2s = 15, g3s = 22;

    // Workspace (floats), reused across the 4 sequential batches (~22 MB)
    float* ws = (float*)d_ws;
    size_t off = 0;
    float* XL   = ws + off; off += (size_t)NN * HIDD;
    float* XR   = ws + off; off += (size_t)NN * HIDD;
    float* HA   = ws + off; off += (size_t)NN * HIDD;
    float* HB   = ws + off; off += (size_t)NN * HIDD;
    float* EMX  = ws + off; off += NN;
    float* DEN  = ws + off; off += NN;
    float* EBUF = ws + off; off += (EE + NN);
    float* H0   = ws + off; off += NN;
    float* MEAN = ws + off; off += 1;

    float* outp = (float*)d_out;

    auto fillF = [&](float* p, float v, int n) {
        fill_kernel<<<CDIV(n, 256), 256, 0, stream>>>(p, v, n);
    };
    const int tot = EE + NN;

    for (int b = 0; b < BB; ++b) {
        const float* xb  = x + (size_t)b * NN * INDIM;
        const int*   src = ei + (size_t)b * 2 * EE;
        const int*   dst = src + EE;
        float* rec = outp + (size_t)b * NN * INDIM;
        float* z   = outp + (size_t)BB * NN * INDIM + (size_t)b * NN * OUTDIM;
        float* eat = outp + (size_t)BB * NN * (INDIM + OUTDIM) + (size_t)b * EE;

        // Encoder MLP: 12 -> 64 (relu) -> 64 (relu) -> 3
        launch_gemm<INDIM, HIDD, 1>(xb, P(2), P(3), XL, NN, stream);
        launch_gemm<HIDD,  HIDD, 1>(XL, P(4), P(5), XR, NN, stream);
        launch_gemm<HIDD, OUTDIM, 0>(XR, P(6), P(7), z, NN, stream);

        // Edge attributes + mean (for self loops)
        fillF(MEAN, 0.f, 1);
        eattr_kernel<<<CDIV(EE, 256), 256, 0, stream>>>(z, src, dst, eat, MEAN);
        extract_h0<<<CDIV(NN, 256), 256, 0, stream>>>(z, H0);

        // ---- GAT layer 1: 1 -> 64, relu ------------------------------------
        launch_gemm<1, HIDD, 0>(H0, P(g1s + oWl), P(g1s + obl), XL, NN, stream);
        launch_gemm<1, HIDD, 0>(H0, P(g1s + oWr), P(g1s + obr), XR, NN, stream);
        fillF(EMX, -INFINITY, NN); fillF(DEN, 0.f, NN); fillF(HA, 0.f, NN * HIDD);
        gat_score<HIDD><<<CDIV(tot, 256), 256, 0, stream>>>(XL, XR, src, dst, eat,
                    MEAN, P(g1s + oWe), P(g1s + oatt), EBUF, EMX);
        gat_weight<<<CDIV(tot, 256), 256, 0, stream>>>(dst, EBUF, EMX, DEN);
        gat_accum<HIDD><<<CDIV(tot, 256), 256, 0, stream>>>(XL, src, dst, EBUF, DEN, HA);
        bias_act_kernel<<<CDIV(NN * HIDD, 256), 256, 0, stream>>>(HA, P(g1s + obias), NN, HIDD, 1);

        // ---- GAT layer 2: 64 -> 64, relu -----------------------------------
        launch_gemm<HIDD, HIDD, 0>(HA, P(g2s + oWl), P(g2s + obl), XL, NN, stream);
        launch_gemm<HIDD, HIDD, 0>(HA, P(g2s + oWr), P(g2s + obr), XR, NN, stream);
        fillF(EMX, -INFINITY, NN); fillF(DEN, 0.f, NN); fillF(HB, 0.f, NN * HIDD);
        gat_score<HIDD><<<CDIV(tot, 256), 256, 0, stream>>>(XL, XR, src, dst, eat,
                    MEAN, P(g2s + oWe), P(g2s + oatt), EBUF, EMX);
        gat_weight<<<CDIV(tot, 256), 256, 0, stream>>>(dst, EBUF, EMX, DEN);
        gat_accum<HIDD><<<CDIV(tot, 256), 256, 0, stream>>>(XL, src, dst, EBUF, DEN, HB);
        bias_act_kernel<<<CDIV(NN * HIDD, 256), 256, 0, stream>>>(HB, P(g2s + obias), NN, HIDD, 1);

        // ---- GAT layer 3: 64 -> 3, no relu -> g in HA (N x 3) --------------
        launch_gemm<HIDD, OUTDIM, 0>(HB, P(g3s + oWl), P(g3s + obl), XL, NN, stream);
        launch_gemm<HIDD, OUTDIM, 0>(HB, P(g3s + oWr), P(g3s + obr), XR, NN, stream);
        fillF(EMX, -INFINITY, NN); fillF(DEN, 0.f, NN); fillF(HA, 0.f, NN * OUTDIM);
        gat_score<OUTDIM><<<CDIV(tot, 256), 256, 0, stream>>>(XL, XR, src, dst, eat,
                    MEAN, P(g3s + oWe), P(g3s + oatt), EBUF, EMX);
        gat_weight<<<CDIV(tot, 256), 256, 0, stream>>>(dst, EBUF, EMX, DEN);
        gat_accum<OUTDIM><<<CDIV(tot, 256), 256, 0, stream>>>(XL, src, dst, EBUF, DEN, HA);
        bias_act_kernel<<<CDIV(NN * OUTDIM, 256), 256, 0, stream>>>(HA, P(g3s + obias), NN, OUTDIM, 0);

        // skip MLP on z: 3 -> 64 (relu) -> 64 (relu) -> 3   (into HB as N x 3)
        launch_gemm<OUTDIM, HIDD, 1>(z,  P(skips),     P(skips + 1), XL, NN, stream);
        launch_gemm<HIDD,   HIDD, 1>(XL, P(skips + 2), P(skips + 3), XR, NN, stream);
        launch_gemm<HIDD, OUTDIM, 0>(XR, P(skips + 4), P(skips + 5), HB, NN, stream);

        // comb = g + alpha * skip   (into XL as N x 3)
        comb_kernel<<<CDIV(NN * OUTDIM, 256), 256, 0, stream>>>(HA, HB, XL, NN * OUTDIM);

        // head MLP: 3 -> 64 (relu) -> 64 (relu) -> 12  -> rec
        launch_gemm<OUTDIM, HIDD, 1>(XL, P(heads),     P(heads + 1), XR,  NN, stream);
        launch_gemm<HIDD,   HIDD, 1>(XR, P(heads + 2), P(heads + 3), HB,  NN, stream);
        launch_gemm<HIDD,  INDIM, 0>(HB, P(heads + 4), P(heads + 5), rec, NN, stream);
    }
}